// GELU59_17566416240689
// MI455X (gfx1250) — compile-verified
//
#include <hip/hip_runtime.h>
#include <hip/hip_bf16.h>
#include <math.h>

// ---------------------------------------------------------------------------
// Fused: gelu -> l2norm -> sims(WMMA f32 16x16x4) -> logsumexp gate -> scale
// B=8, T=2048, D=2048, K=8  (rows = B*T = 16384, tiles of 16 rows per block)
// Memory-bound: one read of x, one write of out. Sims GEMM fused via
// V_WMMA_F32_16X16X4_F32 (full f32 precision, matches the f32 reference).
//
// Workspace prototype buffer has 9 rows: rows 0..7 = normalized prototypes,
// row 8 = zeros. WMMA B-fragment lanes with n>=8 read the zero row, so the
// inner loop has NO divergent branches (previous version emitted
// s_and_saveexec + scalar 4B loads per element).
// ---------------------------------------------------------------------------

typedef __attribute__((ext_vector_type(2))) float v2f;
typedef __attribute__((ext_vector_type(8))) float v8f;

#define D_DIM 2048
#define K_PROTO 8
#define TILE_ROWS 16
#define TSTRIDE 2052            // 2048 + 4 pad: rows -> distinct LDS bank groups
#define EPS 1e-12f
#define SQRT_2_OVER_PI 0.7978845608028654f

__device__ __forceinline__ float gelu_tanh(float v) {
    float v3 = v * v * v;
    return 0.5f * v * (1.0f + tanhf(SQRT_2_OVER_PI * (v + 0.044715f * v3)));
}

// ---------------------------------------------------------------------------
// Kernel 0: normalize the 8 prototype rows into workspace (pn = protos/||p||)
// and zero the 9th (padding) row. Launched with K_PROTO+1 blocks.
// ---------------------------------------------------------------------------
__global__ void __launch_bounds__(256)
proto_norm_kernel(const float* __restrict__ protos, float* __restrict__ pnz) {
    const int k = blockIdx.x;          // 0..8
    const int t = threadIdx.x;
    if (k == K_PROTO) {                // uniform per-block branch: zero row 8
        for (int c = t; c < D_DIM; c += 256) pnz[K_PROTO * D_DIM + c] = 0.0f;
        return;
    }
    __shared__ float red[256];
    float ssq = 0.0f;
    for (int c = t; c < D_DIM; c += 256) {
        float v = protos[k * D_DIM + c];
        ssq += v * v;
    }
    red[t] = ssq;
    __syncthreads();
    for (int s = 128; s > 0; s >>= 1) {
        if (t < s) red[t] += red[t + s];
        __syncthreads();
    }
    const float scale = 1.0f / fmaxf(sqrtf(red[0]), EPS);
    for (int c = t; c < D_DIM; c += 256) {
        pnz[k * D_DIM + c] = protos[k * D_DIM + c] * scale;
    }
}

// ---------------------------------------------------------------------------
// Main fused kernel: one 16-row tile per 256-thread block (8 waves).
// ---------------------------------------------------------------------------
__global__ void __launch_bounds__(256)
fused_gate_kernel(const float* __restrict__ x,
                  const float* __restrict__ pnz,   // 9 rows: 8 protos + zeros
                  const float* __restrict__ p_log_tau,
                  const float* __restrict__ p_log_gamma,
                  const float* __restrict__ p_log_blend,
                  float* __restrict__ out) {
    extern __shared__ float smem[];
    float* tile    = smem;                         // 16 * 2052
    float* partial = tile + TILE_ROWS * TSTRIDE;   // 8 * 256 (per-wave C tiles)
    float* red     = partial + 8 * 256;            // 256
    float* norms   = red + 256;                    // 16
    float* simsbuf = norms + 16;                   // 16 * 8
    float* gate    = simsbuf + 16 * K_PROTO;       // 16

    const int t = threadIdx.x;
    const int m = t >> 4;                  // tile row 0..15
    const int s = t & 15;                  // sub-lane within row
    const long grow = (long)blockIdx.x * TILE_ROWS + m;
    const float* xrow = x + grow * D_DIM;
    float* orow = out + grow * D_DIM;
    float* trow = tile + m * TSTRIDE;

    // ---- Phase 1: gelu(x) -> LDS tile, per-thread partial sum of squares ----
    float ssq = 0.0f;
    #pragma unroll 4
    for (int j = 0; j < 32; ++j) {
        const int col = s * 4 + j * 64;           // 16B-aligned
        float4 v = *reinterpret_cast<const float4*>(xrow + col);
        float4 g;
        g.x = gelu_tanh(v.x);
        g.y = gelu_tanh(v.y);
        g.z = gelu_tanh(v.z);
        g.w = gelu_tanh(v.w);
        *reinterpret_cast<float4*>(trow + col) = g;
        ssq += g.x * g.x + g.y * g.y + g.z * g.z + g.w * g.w;
    }
    red[t] = ssq;
    __syncthreads();

    // ---- Row norms (threads 0..15), in parallel with WMMA phase ------------
    if (t < TILE_ROWS) {
        float acc = 0.0f;
        #pragma unroll
        for (int i = 0; i < 16; ++i) acc += red[t * 16 + i];
        norms[t] = sqrtf(acc);
    }

    // ---- Phase 2: sims_raw[16x8] = tile(16xD) * pn^T(Dx8) via WMMA f32 -----
    // Each wave owns a 256-wide D-chunk: 64 steps of V_WMMA_F32_16X16X4_F32.
    // A(16x4): lane L (m=L&15, hi=L>>4): VGPR0/1 hold K = 2*hi, 2*hi+1
    // B(4x16): lane L (n=L&15): VGPR0/1 hold K = 2*hi, 2*hi+1 ; n>=8 -> zero
    // row. All loads unconditional 8-byte (b64) accesses -> no divergence.
    {
        const int lane = t & 31;
        const int w = t >> 5;                  // wave id 0..7
        const int mn = lane & 15;
        const int hi = lane >> 4;
        const int dbase = w * 256;
        const int bn = (mn < K_PROTO) ? mn : K_PROTO;   // zero row for n>=8
        const float* arow = tile + mn * TSTRIDE;
        const v2f*  brow2 = reinterpret_cast<const v2f*>(pnz + bn * D_DIM);

        v8f acc = {};
        #pragma unroll 4
        for (int j = 0; j < 64; ++j) {
            const int d0 = dbase + j * 4 + 2 * hi;      // even -> 8B aligned
            v2f a = *reinterpret_cast<const v2f*>(arow + d0);
            v2f b = brow2[d0 >> 1];
            acc = __builtin_amdgcn_wmma_f32_16x16x4_f32(
                false, a, false, b, (short)0, acc, false, false);
        }
        // C layout: VGPR v, lane L -> (M = v + 8*hi, N = L&15)
        #pragma unroll
        for (int v = 0; v < 8; ++v) {
            const int M = v + 8 * hi;
            partial[w * 256 + M * 16 + mn] = acc[v];
        }
    }
    __syncthreads();

    // ---- Reduce per-wave partials; divide by row norm -> sims[16x8] --------
    {
        const int M = t >> 4;
        const int N = t & 15;
        float sraw = 0.0f;
        #pragma unroll
        for (int w2 = 0; w2 < 8; ++w2) sraw += partial[w2 * 256 + t];
        if (N < K_PROTO) {
            const float inv = 1.0f / fmaxf(norms[M], EPS);
            simsbuf[M * K_PROTO + N] = sraw * inv;
        }
    }
    __syncthreads();

    // ---- Per-row logsumexp gate (threads 0..15) ----------------------------
    if (t < TILE_ROWS) {
        const float tau   = expf(p_log_tau[0]);
        const float gamma = expf(p_log_gamma[0]);
        const float alpha = 1.0f / (1.0f + expf(-p_log_blend[0]));
        float y[K_PROTO];
        float mx = -INFINITY;
        #pragma unroll
        for (int k = 0; k < K_PROTO; ++k) {
            y[k] = simsbuf[t * K_PROTO + k] * tau;
            mx = fmaxf(mx, y[k]);
        }
        float sum = 0.0f;
        #pragma unroll
        for (int k = 0; k < K_PROTO; ++k) sum += expf(y[k] - mx);
        const float lse = mx + logf(sum);
        const float soft = lse / tau - logf((float)K_PROTO) / tau;
        const float novelty = expf(-gamma * soft);
        gate[t] = 1.0f - alpha + alpha * novelty;
    }
    __syncthreads();

    // ---- Phase 3: scale tile by gate, write out ----------------------------
    const float g = gate[m];
    #pragma unroll 4
    for (int j = 0; j < 32; ++j) {
        const int col = s * 4 + j * 64;
        float4 v = *reinterpret_cast<float4*>(trow + col);
        v.x *= g; v.y *= g; v.z *= g; v.w *= g;
        *reinterpret_cast<float4*>(orow + col) = v;
    }
}

extern "C" void kernel_launch(void* const* d_in, const int* in_sizes, int n_in,
                              void* d_out, int out_size, void* d_ws, size_t ws_size,
                              hipStream_t stream) {
    const float* x          = (const float*)d_in[0];
    const float* protos     = (const float*)d_in[1];
    const float* log_tau    = (const float*)d_in[2];
    const float* log_gamma  = (const float*)d_in[3];
    const float* log_blend  = (const float*)d_in[4];
    float* out = (float*)d_out;
    float* pnz = (float*)d_ws;              // 9 * 2048 floats = 72 KB

    const int rows = in_sizes[0] / D_DIM;   // 16384
    const int tiles = rows / TILE_ROWS;     // 1024

    proto_norm_kernel<<<K_PROTO + 1, 256, 0, stream>>>(protos, pnz);

    constexpr size_t SHMEM =
        (TILE_ROWS * TSTRIDE + 8 * 256 + 256 + 16 + 16 * K_PROTO + 16) *
        sizeof(float);  // ~138 KB (<= 320 KB/WGP; 2 blocks resident per WGP)
    fused_gate_kernel<<<tiles, 256, SHMEM, stream>>>(
        x, pnz, log_tau, log_gamma, log_blend, out);
}